// GINConv_47777216201069
// MI455X (gfx1250) — compile-verified
//
#include <hip/hip_runtime.h>
#include <hip/hip_bf16.h>

#define IN_FEAT  256
#define OUT_FEAT 128

typedef float v2f __attribute__((ext_vector_type(2)));
typedef float v8f __attribute__((ext_vector_type(8)));

// ---------------------------------------------------------------------------
// Kernel 1: wx = x @ W  via V_WMMA_F32_16X16X4_F32 (full fp32 precision).
// 8 waves / block, each wave computes a 16x128 output stripe (8 N-tiles).
// W is staged into LDS (128 KB; CDNA5 WGP has 320 KB) repacked as K-pair
// float2s so every per-lane B fragment is ONE aligned ds_load_b64 that lands
// directly in an even VGPR pair (no pairing moves before the WMMA).
//   wlds2[t * 128 + col] = ( W[2t][col], W[2t+1][col] ),  t = 0..127
// Per-lane fragment for K-chunk k (khalf = lane>>4):
//   B = wlds2[ ((k>>1) + khalf) * 128 + col ]
// ---------------------------------------------------------------------------
__global__ __launch_bounds__(256)
void gin_gemm_wmma(const float* __restrict__ x,
                   const float* __restrict__ w,
                   float* __restrict__ wx,
                   int n_nodes)
{
    __shared__ v2f wlds2[(IN_FEAT / 2) * OUT_FEAT];   // 128 KB

    // Cooperative repack of W into K-paired LDS layout (coalesced reads).
    {
        const int total = (IN_FEAT / 2) * OUT_FEAT;   // 16384
        for (int i = threadIdx.x; i < total; i += blockDim.x) {
            const int t = i >> 7;          // K-pair index (OUT_FEAT == 128)
            const int c = i & 127;         // column
            v2f p;
            p.x = w[(2 * t + 0) * OUT_FEAT + c];
            p.y = w[(2 * t + 1) * OUT_FEAT + c];
            wlds2[i] = p;
        }
    }
    __syncthreads();

    const int wave  = threadIdx.x >> 5;          // 0..7
    const int lane  = threadIdx.x & 31;
    const int mtile = blockIdx.x * 8 + wave;     // 16 rows per tile
    if (mtile * 16 + 15 >= n_nodes) return;      // wave-uniform -> EXEC stays all-1

    const int row   = lane & 15;                 // M (A) / N (B,C,D) position
    const int khalf = lane >> 4;                 // 0: K+{0,1}, 1: K+{2,3}

    // A stream: lane reads x[row][k + khalf*2 + {0,1}] as one b64 load.
    const v2f* __restrict__ apair =
        (const v2f*)(x + (size_t)(mtile * 16 + row) * IN_FEAT + khalf * 2);
    // B stream base: per-lane column + khalf row of the paired layout.
    const v2f* __restrict__ bbase = wlds2 + khalf * OUT_FEAT + row;

    v8f acc[8];
#pragma unroll
    for (int n = 0; n < 8; ++n) acc[n] = (v8f)0.0f;

    for (int k = 0; k < IN_FEAT; k += 4) {
        const v2f a = apair[k >> 1];                       // global b64
        const v2f* __restrict__ bk = bbase + (k >> 1) * OUT_FEAT;
#pragma unroll
        for (int n = 0; n < 8; ++n) {
            const v2f b = bk[n * 16];                      // one ds_load_b64
            acc[n] = __builtin_amdgcn_wmma_f32_16x16x4_f32(
                /*neg_a=*/false, a, /*neg_b=*/false, b,
                /*c_mod=*/(short)0, acc[n],
                /*reuse_a=*/false, /*reuse_b=*/false);
        }
    }

    // C/D layout: VGPR v -> (M = v + khalf*8, N = row)
#pragma unroll
    for (int n = 0; n < 8; ++n) {
#pragma unroll
        for (int v = 0; v < 8; ++v) {
            const int m = v + khalf * 8;
            wx[(size_t)(mtile * 16 + m) * OUT_FEAT + n * 16 + row] = acc[n][v];
        }
    }
}

// ---------------------------------------------------------------------------
// Kernel 2: out = eps * wx   (also the deterministic init of out)
// ---------------------------------------------------------------------------
__global__ __launch_bounds__(256)
void gin_init_eps(const float* __restrict__ wx,
                  const float* __restrict__ eps,
                  float* __restrict__ out,
                  int n4)
{
    const int i = blockIdx.x * blockDim.x + threadIdx.x;
    if (i >= n4) return;
    const float e = eps[0];
    float4 v = ((const float4*)wx)[i];
    float4 r;
    r.x = e * v.x; r.y = e * v.y; r.z = e * v.z; r.w = e * v.w;
    ((float4*)out)[i] = r;
}

// ---------------------------------------------------------------------------
// Kernel 3: edge scatter: out[dst] += wx[src].
// One wave per edge; each lane moves 4 contiguous floats (b128 gather from
// wx, 4x global_atomic_add_f32 into out). wx + out (2 x 51.2 MB) fit in the
// 192 MB L2, so the random gather and the atomic RMWs stay on-chip.
// ---------------------------------------------------------------------------
__global__ __launch_bounds__(256)
void gin_scatter(const float* __restrict__ wx,
                 const int*   __restrict__ src,
                 const int*   __restrict__ dst,
                 float* __restrict__ out,
                 int n_edges)
{
    const int e = blockIdx.x * 8 + (threadIdx.x >> 5);
    if (e >= n_edges) return;
    const int lane = threadIdx.x & 31;

    const int s = src[e];
    const int d = dst[e];

    const float4 v = ((const float4*)(wx + (size_t)s * OUT_FEAT))[lane];
    float* o = out + (size_t)d * OUT_FEAT + lane * 4;
    atomicAdd(o + 0, v.x);
    atomicAdd(o + 1, v.y);
    atomicAdd(o + 2, v.z);
    atomicAdd(o + 3, v.w);
}

// ---------------------------------------------------------------------------
extern "C" void kernel_launch(void* const* d_in, const int* in_sizes, int n_in,
                              void* d_out, int out_size, void* d_ws, size_t ws_size,
                              hipStream_t stream)
{
    const float* x    = (const float*)d_in[0];   // [N, 256] fp32
    const float* w    = (const float*)d_in[1];   // [256, 128] fp32
    const float* eps  = (const float*)d_in[2];   // [1] fp32
    const int*   ei   = (const int*)d_in[3];     // [2, E] int
    float*       out  = (float*)d_out;           // [N, 128] fp32
    float*       wx   = (float*)d_ws;            // [N, 128] fp32 scratch (51.2 MB)

    const int n_nodes = in_sizes[0] / IN_FEAT;
    const int n_edges = in_sizes[3] / 2;

    // 1) GEMM: wx = x @ W
    const int mtiles  = (n_nodes + 15) / 16;
    const int gblocks = (mtiles + 7) / 8;
    gin_gemm_wmma<<<gblocks, 256, 0, stream>>>(x, w, wx, n_nodes);

    // 2) out = eps * wx
    const int n4 = (n_nodes * OUT_FEAT) / 4;
    gin_init_eps<<<(n4 + 255) / 256, 256, 0, stream>>>(wx, eps, out, n4);

    // 3) out[dst] += wx[src] over all edges
    gin_scatter<<<(n_edges + 7) / 8, 256, 0, stream>>>(
        wx, ei, ei + n_edges, out, n_edges);
}